// LocalNeighborhood_1666447311239
// MI455X (gfx1250) — compile-verified
//
#include <hip/hip_runtime.h>

// ---------------------------------------------------------------------------
// LocalNeighborhood on MI455X (gfx1250, wave32)
//   frame: (4, 2048, 4, 3) f32   [center = row 0, axes = rows 1..3]
//   attrs: (4, 2048, 128) f32
//   out  : (4, 2048, 16, 131) f32  = [delta . axes (3) | gathered attrs (128)]
//
// Distance scores via V_WMMA_F32_16X16X4_F32:
//   rank(d^2) == rank(-2 q.c + |c|^2)   (|q|^2 constant per query row)
//   A (16x4, M=query) row  = (qx, qy, qz, 1)
//   B (4x16, N=cand)  col  = (-2cx, -2cy, -2cz, |c|^2)
// HBM roofline: ~69 MB output stores -> ~3us floor @ 23.3 TB/s; attribute
// gathers (4 MB footprint) are L2-resident. Scoring/top-k must stay on-chip.
// ---------------------------------------------------------------------------

#define BATCH   4
#define NPTS    2048
#define KMAX    16
#define QTILE   16
#define CHUNK   512
#define NCHUNK  (NPTS / CHUNK)     // 4
#define ATTR    128
#define OUTW    (3 + ATTR)         // 131

typedef __attribute__((ext_vector_type(2))) float v2f;
typedef __attribute__((ext_vector_type(8))) float v8f;

__global__ __launch_bounds__(256) void local_nbhd_kernel(
    const float* __restrict__ frame,   // (B, L, 4, 3)
    const float* __restrict__ attrs,   // (B, L, 128)
    float* __restrict__ out)           // (B, L, 16, 131)
{
    // 32KB pane: phase B/C = 16x512 f32 score tile; phase D = reused as
    // 16x256 f32 merge-scores + 16x256 i32 merge-indices (same 32KB).
    __shared__ float smem[QTILE * CHUNK];
    __shared__ float qc[QTILE][3];        // query centers
    __shared__ float qax[QTILE][3][3];    // query axes (axis, d)
    __shared__ int   final_idx[QTILE][KMAX];

    const int tid  = threadIdx.x;
    const int lane = tid & 31;            // wave32
    const int wave = tid >> 5;            // 8 waves

    const int b  = blockIdx.x / (NPTS / QTILE);
    const int q0 = (blockIdx.x % (NPTS / QTILE)) * QTILE;

    const float* fbase = frame + (size_t)b * NPTS * 12;   // 4x3 per point
    const float* abase = attrs + (size_t)b * NPTS * ATTR;

    // ---- load the 16-query tile (center + 3 axes) into LDS --------------
    if (tid < QTILE * 12) {
        int m = tid / 12, e = tid % 12;
        float v = fbase[(size_t)(q0 + m) * 12 + e];
        if (e < 3) qc[m][e] = v;
        else       qax[m][e / 3 - 1][e % 3] = v;
    }
    __syncthreads();

    // ---- A operand (16x4 f32, ISA 7.12.2 layout), branchless ------------
    // lanes 0-15:  M=lane,    v0 = K0 = qx, v1 = K1 = qy
    // lanes 16-31: M=lane-16, v0 = K2 = qz, v1 = K3 = 1.0
    const int  mrow   = lane & 15;
    const bool lohalf = (lane < 16);
    v2f amat;
    amat.x = lohalf ? qc[mrow][0] : qc[mrow][2];
    amat.y = lohalf ? qc[mrow][1] : 1.0f;

    // ---- per-thread running top-16 (register compare-exchange cascade) --
    const int q   = tid >> 4;     // query this thread selects for
    const int seg = tid & 15;     // its candidate sub-stripe
    float ts[KMAX]; int ti[KMAX];
#pragma unroll
    for (int j = 0; j < KMAX; ++j) { ts[j] = 3.4e38f; ti[j] = 0x7fffffff; }

    auto fold = [&](float s, int ci) {
        if (s < ts[KMAX - 1]) {
            float cs = s; int cidx = ci;
#pragma unroll
            for (int jj = 0; jj < KMAX; ++jj) {
                bool lt = cs < ts[jj];
                float ns = lt ? ts[jj] : cs;
                int   ni = lt ? ti[jj] : cidx;
                ts[jj] = lt ? cs   : ts[jj];
                ti[jj] = lt ? cidx : ti[jj];
                cs = ns; cidx = ni;
            }
        }
    };

    for (int c = 0; c < NCHUNK; ++c) {
        // -- Phase B: 8 waves x 4 WMMA tiles -> 16x512 score pane in LDS --
#pragma unroll
        for (int t = 0; t < 4; ++t) {
            const int tile = wave * 4 + t;
            const int cand = c * CHUNK + tile * 16 + (lane & 15);
            // one b128 load covers (cx,cy,cz[,next]); 16B-aligned, in-bounds
            const float4 c4 =
                *reinterpret_cast<const float4*>(fbase + (size_t)cand * 12);
            const float cx = c4.x, cy = c4.y, cz = c4.z;
            const float n2 = __builtin_fmaf(cx, cx,
                              __builtin_fmaf(cy, cy, cz * cz));
            // B operand (4x16): lanes 0-15 -> K0,K1 ; lanes 16-31 -> K2,K3
            // branchless -> v_cndmask, keeps EXEC all-1s for WMMA
            v2f bmat;
            bmat.x = -2.0f * (lohalf ? cx : cz);
            bmat.y =  lohalf ? (-2.0f * cy) : n2;

            v8f acc = {};
            acc = __builtin_amdgcn_wmma_f32_16x16x4_f32(
                false, amat, false, bmat, (short)0, acc, false, false);

            // D layout: lanes 0-15 -> M=v, N=lane ; lanes 16-31 -> M=v+8
            const int col  = tile * 16 + (lane & 15);
            const int moff = lohalf ? 0 : 8;
#pragma unroll
            for (int v = 0; v < 8; ++v)
                smem[(moff + v) * CHUNK + col] = acc[v];
        }
        __syncthreads();

        // -- Phase C: fold 32 candidates/thread into running top16.
        // col = j*64 + seg*4 + e : per j the 16 segs touch banks 0..63
        // exactly once (float4 DS loads, 16B aligned, ~conflict-free).
#pragma unroll
        for (int j = 0; j < 8; ++j) {
            const int colbase = j * 64 + seg * 4;
            const float4 s4 = *reinterpret_cast<const float4*>(
                smem + q * CHUNK + colbase);
            const int ci = c * CHUNK + colbase;
            fold(s4.x, ci + 0);
            fold(s4.y, ci + 1);
            fold(s4.z, ci + 2);
            fold(s4.w, ci + 3);
        }
        __syncthreads();   // pane about to be overwritten
    }

    // ---- Phase D: dump segment lists to LDS (pane reuse), merge per query
    float* msc = smem;                       // [16][256] f32
    int*   mix = (int*)(smem + QTILE * 256); // [16][256] i32 (2nd 16KB half)
#pragma unroll
    for (int j = 0; j < KMAX; ++j) {
        msc[q * 256 + seg * KMAX + j] = ts[j];
        mix[q * 256 + seg * KMAX + j] = ti[j];
    }
    __syncthreads();

    if (tid < QTILE) {
        float bs[KMAX]; int bi[KMAX];
#pragma unroll
        for (int j = 0; j < KMAX; ++j) { bs[j] = 3.4e38f; bi[j] = 0x7fffffff; }
        for (int e = 0; e < 256; ++e) {
            float cs = msc[tid * 256 + e];
            int cidx = mix[tid * 256 + e];
            if (cs < bs[KMAX - 1]) {
#pragma unroll
                for (int jj = 0; jj < KMAX; ++jj) {
                    bool lt = cs < bs[jj];
                    float ns = lt ? bs[jj] : cs;
                    int   ni = lt ? bi[jj] : cidx;
                    bs[jj] = lt ? cs   : bs[jj];
                    bi[jj] = lt ? cidx : bi[jj];
                    cs = ns; cidx = ni;
                }
            }
        }
#pragma unroll
        for (int j = 0; j < KMAX; ++j) final_idx[tid][j] = bi[j];
    }
    __syncthreads();

    // ---- Phase E: cooperative output. One wave per (query, k) row:
    // 131 floats = [delta.axes x3 | attrs x128], lane-parallel & coalesced.
    for (int r = wave; r < QTILE * KMAX; r += 8) {
        const int rq = r >> 4, rk = r & 15;
        const int n  = final_idx[rq][rk];
        const float* cp = fbase + (size_t)n * 12;
        const float dx = cp[0] - qc[rq][0];
        const float dy = cp[1] - qc[rq][1];
        const float dz = cp[2] - qc[rq][2];
        const size_t rowbase =
            ((((size_t)b * NPTS + q0 + rq) * KMAX) + rk) * OUTW;
        const float* arow = abase + (size_t)n * ATTR;
        for (int j = lane; j < OUTW; j += 32) {
            float v;
            if (j < 3)
                v = dx * qax[rq][j][0] + dy * qax[rq][j][1] + dz * qax[rq][j][2];
            else
                v = arow[j - 3];
            out[rowbase + j] = v;
        }
    }
}

extern "C" void kernel_launch(void* const* d_in, const int* in_sizes, int n_in,
                              void* d_out, int out_size, void* d_ws, size_t ws_size,
                              hipStream_t stream) {
    const float* frame = (const float*)d_in[0];
    const float* attrs = (const float*)d_in[1];
    float* out = (float*)d_out;

    dim3 grid(BATCH * (NPTS / QTILE));   // 512 blocks (16 queries each)
    dim3 block(256);                     // 8 wave32 waves
    hipLaunchKernelGGL(local_nbhd_kernel, grid, block, 0, stream,
                       frame, attrs, out);
}